// ProbabilityNoiser_50062138802810
// MI455X (gfx1250) — compile-verified
//
#include <hip/hip_runtime.h>

typedef __attribute__((ext_vector_type(2))) float v2f;
typedef __attribute__((ext_vector_type(8))) float v8f;

#define NCLS 512
#define NBATCH 4096
#define BM 32   // batch rows per workgroup (2 row-blocks of 16)

// ---------------------------------------------------------------------------
// Kernel 0: pack V into WMMA-B-operand-friendly layouts (runs once, 2 MB).
//   VPv[(k>>1)*1024 + col*2 + (k&1)] = V [k][col]      (phase-1 B = V)
//   VPt[(k>>1)*1024 + col*2 + (k&1)] = V^T[k][col] = V[col][k]  (phase-2 B)
// With this layout a lane's B pair (K=ka, ka+1) is ONE coalesced b64 load.
// ---------------------------------------------------------------------------
__global__ __launch_bounds__(256) void pack_operands(const float* __restrict__ V,
                                                     float* __restrict__ VPv,
                                                     float* __restrict__ VPt) {
    const int idx = blockIdx.x * 256 + threadIdx.x;  // 0 .. 512*256-1
    const int col = idx & (NCLS - 1);
    const int k2  = idx >> 9;                        // k-pair index, 0..255

    v2f a;
    a.x = V[(size_t)(2 * k2) * NCLS + col];
    a.y = V[(size_t)(2 * k2 + 1) * NCLS + col];
    *(v2f*)(VPv + ((size_t)k2 * NCLS + col) * 2) = a;

    v2f b = *(const v2f*)(V + (size_t)col * NCLS + 2 * k2);  // V[col][2k2..+1]
    *(v2f*)(VPt + ((size_t)k2 * NCLS + col) * 2) = b;
}

// ---------------------------------------------------------------------------
// Fused kernel:  out = ( exp(lam*sigma) * (P0 @ V) ) @ V^T
// One WG = 32 batch rows; 8 waves. Each wave owns BOTH 16-row blocks and a
// 64-column slice (2x4 = 8 f32 16x16 accumulator tiles), so every B operand
// register pair feeds two v_wmma_f32_16x16x4_f32 ops. The scaled phase-1
// result lives in exactly-64KB XOR-swizzled LDS (phys = col ^ ((row&15)<<2))
// giving conflict-free b64 reads in phase 2 without padding.
// ---------------------------------------------------------------------------
__global__ __launch_bounds__(256) void fused_probability_noiser(
    const float* __restrict__ P0,     // [4096,512]
    const float* __restrict__ sigma,  // [4096]
    const float* __restrict__ lam,    // [512]
    const float* __restrict__ VPv,    // packed V
    const float* __restrict__ VPt,    // packed V^T
    float* __restrict__ out)          // [4096,512]
{
    __shared__ float su[BM][NCLS];    // exactly 64 KB

    const int tid  = threadIdx.x;
    const int wave = tid >> 5;        // 0..7
    const int lane = tid & 31;
    const int nn   = lane & 15;       // column-within-tile / A-row
    const int hi   = lane >> 4;       // K-half selector

    const int cbas = wave * 64;                    // 4 col tiles per wave
    const int row0 = blockIdx.x * BM;              // global batch row base
    const int m    = nn;                           // A-matrix row this lane owns

    v8f acc[2][4];
#pragma unroll
    for (int rb = 0; rb < 2; ++rb)
#pragma unroll
        for (int t = 0; t < 4; ++t)
#pragma unroll
            for (int i = 0; i < 8; ++i) acc[rb][t][i] = 0.0f;

    // -------- Phase 1: U = P0_tile @ V --------
    {
        const float* Ap0 = P0 + (size_t)(row0 + m) * NCLS;       // row block 0
        const float* Ap1 = P0 + (size_t)(row0 + 16 + m) * NCLS;  // row block 1
        for (int k = 0; k < NCLS; k += 4) {
            const int ka = k + 2 * hi;                   // this lane's K pair
            v2f a0 = *(const v2f*)(Ap0 + ka);
            v2f a1 = *(const v2f*)(Ap1 + ka);
            const float* Bp =
                VPv + (size_t)((k >> 1) + hi) * (2 * NCLS) + (size_t)(cbas + nn) * 2;
#pragma unroll
            for (int t = 0; t < 4; ++t) {
                v2f b = *(const v2f*)(Bp + t * 32);      // (V[ka][col],V[ka+1][col])
                acc[0][t] = __builtin_amdgcn_wmma_f32_16x16x4_f32(
                    false, a0, false, b, (short)0, acc[0][t], false, false);
                acc[1][t] = __builtin_amdgcn_wmma_f32_16x16x4_f32(
                    false, a1, false, b, (short)0, acc[1][t], false, false);
            }
        }
    }

    // -------- scale by exp(lam*sigma), stash in swizzled LDS --------
    {
        float sg[2][8];
#pragma unroll
        for (int rb = 0; rb < 2; ++rb)
#pragma unroll
            for (int g = 0; g < 8; ++g)
                sg[rb][g] = sigma[row0 + rb * 16 + g + 8 * hi];
#pragma unroll
        for (int t = 0; t < 4; ++t) {
            const int col = cbas + t * 16 + nn;
            const float lv = lam[col];
#pragma unroll
            for (int rb = 0; rb < 2; ++rb)
#pragma unroll
                for (int g = 0; g < 8; ++g) {
                    const int mr = g + 8 * hi;           // row within 16-block
                    const float sc = __expf(lv * sg[rb][g]);
                    su[rb * 16 + mr][col ^ (mr << 2)] = sc * acc[rb][t][g];
                }
        }
    }
    __syncthreads();

    // -------- Phase 2: out = SU @ V^T --------
#pragma unroll
    for (int rb = 0; rb < 2; ++rb)
#pragma unroll
        for (int t = 0; t < 4; ++t)
#pragma unroll
            for (int i = 0; i < 8; ++i) acc[rb][t][i] = 0.0f;

    {
        const float* A0 = &su[m][0];
        const float* A1 = &su[16 + m][0];
        for (int k = 0; k < NCLS; k += 4) {
            const int ka = k + 2 * hi;
            const int sw = ka ^ (m << 2);                // pair stays adjacent
            v2f a0 = *(const v2f*)(A0 + sw);
            v2f a1 = *(const v2f*)(A1 + sw);
            const float* Bp =
                VPt + (size_t)((k >> 1) + hi) * (2 * NCLS) + (size_t)(cbas + nn) * 2;
#pragma unroll
            for (int t = 0; t < 4; ++t) {
                v2f b = *(const v2f*)(Bp + t * 32);
                acc[0][t] = __builtin_amdgcn_wmma_f32_16x16x4_f32(
                    false, a0, false, b, (short)0, acc[0][t], false, false);
                acc[1][t] = __builtin_amdgcn_wmma_f32_16x16x4_f32(
                    false, a1, false, b, (short)0, acc[1][t], false, false);
            }
        }
    }

    // -------- coalesced store of D tiles --------
#pragma unroll
    for (int t = 0; t < 4; ++t) {
        const int col = cbas + t * 16 + nn;
#pragma unroll
        for (int rb = 0; rb < 2; ++rb)
#pragma unroll
            for (int g = 0; g < 8; ++g) {
                const int r = row0 + rb * 16 + g + 8 * hi;
                out[(size_t)r * NCLS + col] = acc[rb][t][g];
            }
    }
}

// ---------------------------------------------------------------------------
extern "C" void kernel_launch(void* const* d_in, const int* in_sizes, int n_in,
                              void* d_out, int out_size, void* d_ws, size_t ws_size,
                              hipStream_t stream) {
    const float* p0    = (const float*)d_in[0];   // [4096,512]
    const float* sigma = (const float*)d_in[1];   // [4096]
    const float* evals = (const float*)d_in[2];   // [512]
    const float* evecs = (const float*)d_in[3];   // [512,512]
    float* out = (float*)d_out;
    float* VPv = (float*)d_ws;                    // 1 MB
    float* VPt = VPv + (size_t)NCLS * NCLS;       // 1 MB

    pack_operands<<<(NCLS * NCLS / 2) / 256, 256, 0, stream>>>(evecs, VPv, VPt);

    fused_probability_noiser<<<NBATCH / BM, 256, 0, stream>>>(
        p0, sigma, evals, VPv, VPt, out);
}